// ResampleFromUV_56410100465681
// MI455X (gfx1250) — compile-verified
//
#include <hip/hip_runtime.h>

// ResampleFromUV for MI455X (gfx1250).
// x: (1, 64, 20, 256, 256) f32, quad_idx: (512,1024) int, tex_uv: (512,1024,2) f32
// out: (1, 64, 512, 1024) f32
//
// Strategy: per-pixel coords/weights computed once into registers, then the
// 64-channel loop is innermost so each channel's 5.24 MB texture slab gets
// ~25x L2 reuse across the 512K random gathers. Non-temporal stores/loads
// (gfx1250 TH=NT) keep the write-once output and read-once coords from
// evicting the texture working set from L2. x0/x1 texels are contiguous
// (u in [0,255) guarantees x1=x0+1 in range), so each row tap is a single
// global_load_b64.

constexpr int TEXH = 256;
constexpr int TEXW = 256;
constexpr int NTEX = 20;
constexpr int NCH  = 64;
constexpr int OH   = 512;
constexpr int OW   = 1024;
constexpr int OHW  = OH * OW;                 // 524288 output pixels
constexpr int CHS  = NTEX * TEXH * TEXW;      // per-channel element stride (1310720)

__global__ __launch_bounds__(256)
void resample_uv_kernel(const float* __restrict__ x,
                        const int*   __restrict__ qidx,
                        const float* __restrict__ uv,
                        float*       __restrict__ out)
{
    const int pix = blockIdx.x * 256 + threadIdx.x;   // grid sized exactly to OHW

    // ---- one-time per-pixel coordinate math (read-once -> non-temporal) ----
    const float u = __builtin_nontemporal_load(uv + 2 * (size_t)pix + 0);
    const float v = __builtin_nontemporal_load(uv + 2 * (size_t)pix + 1);
    const int   n = __builtin_nontemporal_load(qidx + pix);

    const float fx0 = floorf(u);
    const float fy0 = floorf(v);
    const float wx  = u - fx0;
    const float wy  = v - fy0;

    int x0 = (int)fx0;
    int y0 = (int)fy0;
    x0 = min(max(x0, 0), TEXW - 1);               // matches reference clamp
    y0 = min(max(y0, 0), TEXH - 1);
    const int y1 = min(y0 + 1, TEXH - 1);
    // u in [0, 255) => x0 <= 254 => x1 = x0+1 always in range: row tap is a
    // contiguous float2 at (y, x0).

    const float w00 = (1.0f - wx) * (1.0f - wy);
    const float w01 = wx * (1.0f - wy);
    const float w10 = (1.0f - wx) * wy;
    const float w11 = wx * wy;

    // Element offsets within one channel slab (fits in 32 bits: < 1.31M).
    const int off0 = (n * TEXH + y0) * TEXW + x0;  // row y0, cols x0..x0+1
    const int off1 = (n * TEXH + y1) * TEXW + x0;  // row y1, cols x0..x0+1

    // ---- channel loop: 2 x global_load_b64 gathers + 4 FMA + 1 NT store ----
    #pragma unroll 8
    for (int c = 0; c < NCH; ++c) {
        const float* xc = x + (size_t)c * CHS;     // uniform base -> SGPR saddr
        const float2 t0 = *(const float2*)(xc + off0);
        const float2 t1 = *(const float2*)(xc + off1);
        float r = t0.x * w00;
        r = fmaf(t0.y, w01, r);
        r = fmaf(t1.x, w10, r);
        r = fmaf(t1.y, w11, r);
        // Write-once output: non-temporal so it does not evict texture slabs.
        __builtin_nontemporal_store(r, out + (size_t)c * OHW + pix);
    }
}

extern "C" void kernel_launch(void* const* d_in, const int* in_sizes, int n_in,
                              void* d_out, int out_size, void* d_ws, size_t ws_size,
                              hipStream_t stream)
{
    (void)in_sizes; (void)n_in; (void)out_size; (void)d_ws; (void)ws_size;
    const float* x    = (const float*)d_in[0];
    const int*   qidx = (const int*)d_in[1];
    const float* uv   = (const float*)d_in[2];
    float*       out  = (float*)d_out;

    const int threads = 256;                 // 8 wave32s per block
    const int blocks  = OHW / threads;       // 2048 blocks, exact fit
    resample_uv_kernel<<<blocks, threads, 0, stream>>>(x, qidx, uv, out);
}